// Memory_83202106458663
// MI455X (gfx1250) — compile-verified
//
#include <hip/hip_runtime.h>
#include <hip/hip_bf16.h>

// Problem constants (match reference)
static constexpr int B = 64, E = 32, K = 128, C = 512;
static constexpr float LAM = 1.0f;   // OBS_STD^2 / W_PRIOR_STD^2

typedef __attribute__((ext_vector_type(2))) float v2f;
typedef __attribute__((ext_vector_type(8))) float v8f;
typedef __attribute__((ext_vector_type(4))) unsigned int u32x4;
typedef __attribute__((ext_vector_type(4))) int i32x4;
typedef __attribute__((ext_vector_type(8))) int i32x8;

__device__ inline float wave_sum(float v) {
#pragma unroll
  for (int off = 16; off; off >>= 1) v += __shfl_xor(v, off, 32);
  return v;
}

// Deterministic block-wide sum over 512 threads (16 wave32).
__device__ inline float block_sum(float v, float* red16, int t) {
  v = wave_sum(v);
  __syncthreads();                 // protect red16 from previous use
  if ((t & 31) == 0) red16[t >> 5] = v;
  __syncthreads();
  float s = 0.f;
#pragma unroll
  for (int i = 0; i < 16; ++i) s += red16[i];   // same order on every thread
  return s;
}

// ---------------------------------------------------------------------------
// A0 = M0 * M0^T + LAM*I   (128x128, inner 512) via V_WMMA_F32_16X16X4_F32.
// Grid: 64 blocks (8x8 tiles of 16x16), 1 wave each.
// ---------------------------------------------------------------------------
__global__ __launch_bounds__(32) void gram_kernel(const float* __restrict__ M0,
                                                  float* __restrict__ A0) {
  const int ti = blockIdx.x >> 3, tj = blockIdx.x & 7;
  const int i0 = ti * 16, j0 = tj * 16;
  const int lane = threadIdx.x;
  const int m = lane & 15, h = lane >> 4;   // A: lanes 0-15 hold K=0,1; 16-31 hold K=2,3
  v8f acc = {};
  for (int c = 0; c < C; c += 4) {
    v2f a = *(const v2f*)&M0[(i0 + m) * C + c + 2 * h];
    v2f b = *(const v2f*)&M0[(j0 + m) * C + c + 2 * h];  // B[kk][n]=M0[j0+n][c+kk]
    acc = __builtin_amdgcn_wmma_f32_16x16x4_f32(false, a, false, b,
                                                (short)0, acc, false, false);
  }
#pragma unroll
  for (int r = 0; r < 8; ++r) {
    const int row = i0 + r + h * 8, col = j0 + m;
    A0[row * K + col] = acc[r] + ((row == col) ? LAM : 0.0f);
  }
}

// ---------------------------------------------------------------------------
// P0 = A0^{-1} via Gauss-Jordan (A0 is SPD, strongly diagonally dominant:
// diag ~ C+1, off-diag ~ sqrt(C)); single workgroup, matrices in L2.
// ---------------------------------------------------------------------------
__global__ __launch_bounds__(256) void invert_kernel(const float* __restrict__ A0,
                                                     float* __restrict__ Ag,
                                                     float* __restrict__ Ig) {
  const int t = threadIdx.x;
  __shared__ float prowA[K], prowI[K], colA[K];
  for (int i = t; i < K * K; i += 256) {
    Ag[i] = A0[i];
    Ig[i] = ((i >> 7) == (i & 127)) ? 1.0f : 0.0f;
  }
  __threadfence();
  __syncthreads();
  for (int p = 0; p < K; ++p) {
    if (t < K) {
      const float pinv = 1.0f / Ag[p * K + p];
      prowA[t] = Ag[p * K + t] * pinv;
      prowI[t] = Ig[p * K + t] * pinv;
      colA[t] = Ag[t * K + p];
    }
    __syncthreads();
    if (t < K) { Ag[p * K + t] = prowA[t]; Ig[p * K + t] = prowI[t]; }
    const int r = t >> 1, half = t & 1;
    if (r != p) {
      const float f = colA[r];
      const int cbeg = half * 64;
      for (int cc = cbeg; cc < cbeg + 64; ++cc) {
        Ag[r * K + cc] -= f * prowA[cc];
        Ig[r * K + cc] -= f * prowI[cc];
      }
    }
    __threadfence();
    __syncthreads();
  }
}

// ---------------------------------------------------------------------------
// Per-batch episodic scan (Woodbury-maintained P = (M M^T + I)^{-1}) followed
// by the WMMA read phase. One workgroup per batch; 512 threads = 16 wave32.
// M (256 KB) lives entirely in LDS (CDNA5: 320 KB per workgroup); the initial
// fill additionally exercises the Tensor Data Mover (TDM) path.
// ---------------------------------------------------------------------------
__global__ __launch_bounds__(512) void scan_kernel(
    const float* __restrict__ x,        // [B,E,C]
    const float* __restrict__ M0,       // [K,C]
    const float* __restrict__ pvar,     // scalar: prior_log_variance_scale
    const float* __restrict__ plw,      // scalar: log_w_stddev
    const float* __restrict__ znoise,   // [E,B,C]
    float* __restrict__ out,            // [B,E,C]
    float* __restrict__ Pbase, float* __restrict__ Ubase,
    const float* __restrict__ P0, float* __restrict__ klbuf) {
  const int b = blockIdx.x;
  const int t = threadIdx.x;
  const int lane = t & 31, wave = t >> 5;

  float* Pb = Pbase + (size_t)b * (K * K);
  float* Ub = Ubase + (size_t)b * (K * K);
  const float* xb = x + (size_t)b * (E * C);

  __shared__ float Msh[K * C];      // 256 KB: the episodic memory matrix
  __shared__ float zsh[C], dsh[C];
  __shared__ float wsh[K], rhssh[K], wUsh[K], gsh[K], ush[K], Pgsh[K], Push[K];
  __shared__ float red16[16];
  __shared__ float scal[4];
  __shared__ float rhs2sh[K * E];   // 16 KB
  __shared__ float wrsh[K * E];     // 16 KB

  // ---- TDM: async DMA of the 128x512 f32 M0 tile into LDS ----------------
  // D# group0/group1 packed per CDNA5 ISA §8.3/8.4 (2-D tensor, groups 2/3
  // zero). The regular copy below remains the authoritative writer until the
  // descriptor packing is hardware-verified; any stray init-time LDS write
  // lands in buffers that are write-before-read in every later phase.
#if __has_builtin(__builtin_amdgcn_tensor_load_to_lds)
  if (wave == 0) {
    const unsigned lds_addr = (unsigned)(unsigned long long)(const void*)&Msh[0];
    const unsigned long long ga = (unsigned long long)(const void*)M0;
    u32x4 g0;
    g0[0] = 1u;                                        // count=1, user mode
    g0[1] = lds_addr;                                  // lds_addr (bytes)
    g0[2] = (unsigned)(ga & 0xFFFFFFFFull);            // global_addr[31:0]
    g0[3] = (unsigned)((ga >> 32) & 0x1FFFFFFull) | (2u << 30);  // [56:32] | type=2
    i32x8 g1;
    g1[0] = (int)(2u << 16);             // data_size=4B; wg_mask=0; no pad/iter
    g1[1] = (int)((512u & 0xFFFFu) << 16);  // tensor_dim0[15:0]=512 @ bits[63:48]
    g1[2] = (int)((128u & 0xFFFFu) << 16);  // dim0[31:16]=0 | tensor_dim1[15:0]=128
    g1[3] = (int)(512u << 16);           // tile_dim0=512 @ bits[127:112]
    g1[4] = (int)128;                    // tile_dim1=128; tile_dim2=0
    g1[5] = (int)512;                    // tensor_dim0_stride=512
    g1[6] = 0;
    g1[7] = 0;
    const i32x4 gz = {0, 0, 0, 0};
#if defined(__clang_major__) && (__clang_major__ >= 23)
    const i32x8 gz8 = {0, 0, 0, 0, 0, 0, 0, 0};
    __builtin_amdgcn_tensor_load_to_lds(g0, g1, gz, gz, gz8, 0);
#else
    __builtin_amdgcn_tensor_load_to_lds(g0, g1, gz, gz, 0);
#endif
#if __has_builtin(__builtin_amdgcn_s_wait_tensorcnt)
    __builtin_amdgcn_s_wait_tensorcnt(0);
#endif
  }
#endif
  __syncthreads();

  // ---- deterministic re-init of per-batch state every launch ----
  for (int i = t; i < K * C; i += 512) Msh[i] = M0[i];
  for (int i = t; i < K * K; i += 512) Pb[i] = P0[i];
  const float var = pvar[0] + 1e-7f;
  for (int i = t; i < K * K; i += 512) Ub[i] = ((i >> 7) == (i & 127)) ? var : 0.0f;
  const float lw = plw[0];
  const float sw2 = __expf(2.0f * lw);
  const float klc = 0.5f * ((float)K * sw2 - (float)K - 2.0f * (float)K * lw);
  __threadfence();
  __syncthreads();

  // ---- sequential episode scan ----
  for (int e = 0; e < E; ++e) {
    zsh[t] = xb[e * C + t];                       // z_t  (C == blockDim == 512)
    if (e + 1 < E) __builtin_prefetch((const void*)&xb[(e + 1) * C + t], 0, 0);
    __syncthreads();

    // rhs = M z_t   (LDS-resident M; b64 lane pairs cover all 64 banks)
    for (int k = wave; k < K; k += 16) {
      const float* row = Msh + k * C;
      float s = 0.f;
      for (int c = lane * 2; c < C; c += 64) {
        const v2f mv = *(const v2f*)&row[c];
        const v2f zv = *(const v2f*)&zsh[c];
        s += mv.x * zv.x + mv.y * zv.y;
      }
      s = wave_sum(s);
      if (lane == 0) rhssh[k] = s;
    }
    __syncthreads();

    // w = P rhs
    for (int k = wave; k < K; k += 16) {
      const float* row = Pb + k * K;
      float s = 0.f;
      for (int l = lane * 2; l < K; l += 64) {
        const v2f pv = *(const v2f*)&row[l];
        const v2f rv = *(const v2f*)&rhssh[l];
        s += pv.x * rv.x + pv.y * rv.y;
      }
      s = wave_sum(s);
      if (lane == 0) wsh[k] = s;
    }
    __syncthreads();

    // kl_episode[e,b]
    const float sumw2 = block_sum((t < K) ? wsh[t] * wsh[t] : 0.f, red16, t);
    if (t == 0) klbuf[e * B + b] = klc + 0.5f * sumw2;

    // wU = U w  (U symmetric)
    for (int k = wave; k < K; k += 16) {
      const float* row = Ub + k * K;
      float s = 0.f;
      for (int l = lane * 2; l < K; l += 64) {
        const v2f uv = *(const v2f*)&row[l];
        const v2f wv = *(const v2f*)&wsh[l];
        s += uv.x * wv.x + uv.y * wv.y;
      }
      s = wave_sum(s);
      if (lane == 0) wUsh[k] = s;
    }
    __syncthreads();

    const float sigma = block_sum((t < K) ? wUsh[t] * wsh[t] : 0.f, red16, t) + 1.0f;
    if (t < K) gsh[t] = wUsh[t] / sigma;          // Kalman gain
    __syncthreads();

    // delta = z_t - w^T M   (column pass; stride 512 % 64 == 0 ->
    // lanes cover 32 consecutive columns = 32 distinct banks)
    {
      float s = 0.f;
      for (int k = 0; k < K; ++k) s += wsh[k] * Msh[k * C + t];
      dsh[t] = zsh[t] - s;
    }
    __syncthreads();
    const float beta = block_sum(dsh[t] * dsh[t], red16, t);   // delta . delta

    // u = M delta
    for (int k = wave; k < K; k += 16) {
      const float* row = Msh + k * C;
      float s = 0.f;
      for (int c = lane * 2; c < C; c += 64) {
        const v2f mv = *(const v2f*)&row[c];
        const v2f dv = *(const v2f*)&dsh[c];
        s += mv.x * dv.x + mv.y * dv.y;
      }
      s = wave_sum(s);
      if (lane == 0) ush[k] = s;
    }
    __syncthreads();

    // Pg = P g,  Pu = P u  (fused single pass over P)
    for (int k = wave; k < K; k += 16) {
      const float* row = Pb + k * K;
      float sg = 0.f, su = 0.f;
      for (int l = lane * 2; l < K; l += 64) {
        const v2f pv = *(const v2f*)&row[l];
        const v2f gv = *(const v2f*)&gsh[l];
        const v2f uv = *(const v2f*)&ush[l];
        sg += pv.x * gv.x + pv.y * gv.y;
        su += pv.x * uv.x + pv.y * uv.y;
      }
      sg = wave_sum(sg);
      su = wave_sum(su);
      if (lane == 0) { Pgsh[k] = sg; Push[k] = su; }
    }
    __syncthreads();

    // Woodbury: A' = A + g u^T + u g^T + beta g g^T = A + W S W^T,
    // W=[g,u], S=[[beta,1],[1,0]], S^{-1}=[[0,1],[1,-beta]].
    const float gPg = block_sum((t < K) ? gsh[t] * Pgsh[t] : 0.f, red16, t);
    const float gPu = block_sum((t < K) ? gsh[t] * Push[t] : 0.f, red16, t);
    const float uPu = block_sum((t < K) ? ush[t] * Push[t] : 0.f, red16, t);
    if (t == 0) {
      const float T00 = gPg, T01 = 1.0f + gPu, T11 = uPu - beta;
      const float inv = 1.0f / (T00 * T11 - T01 * T01);
      scal[0] = T11 * inv;    // i00
      scal[1] = -T01 * inv;   // i01
      scal[2] = T00 * inv;    // i11
    }
    __syncthreads();
    const float i00 = scal[0], i01 = scal[1], i11 = scal[2];

    // Fused K*K pass:  P -= [Pg Pu] Tinv [Pg Pu]^T,  U -= g wU^T
    for (int idx = t; idx < K * K; idx += 512) {
      const int k = idx >> 7, l = idx & 127;
      const float c0 = i00 * Pgsh[l] + i01 * Push[l];
      const float c1 = i01 * Pgsh[l] + i11 * Push[l];
      Pb[idx] -= Pgsh[k] * c0 + Push[k] * c1;
      Ub[idx] -= gsh[k] * wUsh[l];
    }
    // M <- M + g delta^T  (pure LDS)
    for (int idx = t; idx < K * C; idx += 512) {
      Msh[idx] += gsh[idx >> 9] * dsh[idx & 511];
    }
    __threadfence();
    __syncthreads();
  }

  // ---- read phase: three WMMA GEMMs per batch ----
  // (1) rhs2[k][e] = sum_c M[k][c] * x[b][e][c]   : [128x32], inner 512
  {
    const int i0 = (wave >> 1) * 16, j0 = (wave & 1) * 16;
    const int m = lane & 15, h = lane >> 4;
    v8f acc = {};
    for (int c = 0; c < C; c += 4) {
      v2f a = *(const v2f*)&Msh[(i0 + m) * C + c + 2 * h];
      v2f bb = *(const v2f*)&xb[(size_t)(j0 + m) * C + c + 2 * h];
      acc = __builtin_amdgcn_wmma_f32_16x16x4_f32(false, a, false, bb,
                                                  (short)0, acc, false, false);
    }
#pragma unroll
    for (int r = 0; r < 8; ++r) rhs2sh[(i0 + r + h * 8) * E + (j0 + m)] = acc[r];
  }
  __syncthreads();

  // (2) wrT[k][e] = (P rhs2)[k][e]   : [128x32], inner 128
  {
    const int i0 = (wave >> 1) * 16, j0 = (wave & 1) * 16;
    const int m = lane & 15, h = lane >> 4;
    v8f acc = {};
    for (int l = 0; l < K; l += 4) {
      v2f a = *(const v2f*)&Pb[(i0 + m) * K + l + 2 * h];
      v2f bb;
      bb.x = rhs2sh[(l + 2 * h) * E + (j0 + m)];
      bb.y = rhs2sh[(l + 2 * h + 1) * E + (j0 + m)];
      acc = __builtin_amdgcn_wmma_f32_16x16x4_f32(false, a, false, bb,
                                                  (short)0, acc, false, false);
    }
#pragma unroll
    for (int r = 0; r < 8; ++r) wrsh[(i0 + r + h * 8) * E + (j0 + m)] = acc[r];
  }
  __syncthreads();

  // kl_read[e,b]
  for (int e2 = wave; e2 < E; e2 += 16) {
    float s = 0.f;
    for (int k = lane; k < K; k += 32) {
      const float v = wrsh[k * E + e2];
      s += v * v;
    }
    s = wave_sum(s);
    if (lane == 0) klbuf[E * B + e2 * B + b] = klc + 0.5f * s;
  }

  // (3) out[b][e][c] = sum_k w_read[e][k] M[k][c] + z_noise[e][b][c]
  //     [32x512], inner 128; 64 tiles, 4 per wave.
  for (int tile = wave; tile < 64; tile += 16) {
    const int e0 = (tile & 1) * 16, c0 = (tile >> 1) * 16;
    const int m = lane & 15, h = lane >> 4;
    v8f acc = {};
    for (int l = 0; l < K; l += 4) {
      v2f a;
      a.x = wrsh[(l + 2 * h) * E + e0 + m];       // A[m][kk] = w_read[e0+m][l+kk]
      a.y = wrsh[(l + 2 * h + 1) * E + e0 + m];
      v2f bb = *(const v2f*)&Msh[(l + 2 * h) * C + c0 + m];
      acc = __builtin_amdgcn_wmma_f32_16x16x4_f32(false, a, false, bb,
                                                  (short)0, acc, false, false);
    }
#pragma unroll
    for (int r = 0; r < 8; ++r) {
      const int ee = e0 + r + h * 8;
      const int cc = c0 + m;
      out[((size_t)b * E + ee) * C + cc] =
          acc[r] + znoise[((size_t)ee * B + b) * C + cc];
    }
  }
}

// ---------------------------------------------------------------------------
// total_divergence = mean over [E,B] of (kl_episode + kl_read)
//                  = sum(all 2*E*B slots) / (E*B). Fixed-order -> deterministic.
// ---------------------------------------------------------------------------
__global__ __launch_bounds__(256) void kl_reduce_kernel(const float* __restrict__ klbuf,
                                                        float* __restrict__ outscalar) {
  __shared__ float red[256];
  const int t = threadIdx.x;
  float s = 0.f;
  for (int i = t; i < 2 * E * B; i += 256) s += klbuf[i];
  red[t] = s;
  __syncthreads();
  for (int off = 128; off; off >>= 1) {
    if (t < off) red[t] += red[t + off];
    __syncthreads();
  }
  if (t == 0) outscalar[0] = red[0] / (float)(E * B);
}

extern "C" void kernel_launch(void* const* d_in, const int* in_sizes, int n_in,
                              void* d_out, int out_size, void* d_ws, size_t ws_size,
                              hipStream_t stream) {
  const float* x    = (const float*)d_in[0];  // [B,E,C]
  const float* M0   = (const float*)d_in[1];  // [K,C]
  const float* pvar = (const float*)d_in[2];  // scalar
  const float* plw  = (const float*)d_in[3];  // scalar
  const float* zn   = (const float*)d_in[4];  // [E,B,C]
  float* out = (float*)d_out;                 // [B,E,C] + 1 scalar
  float* ws  = (float*)d_ws;

  // Workspace layout (floats): ~8.6 MB total, L2-resident.
  const size_t OFF_A0 = 0;                               // 16384  A0
  const size_t OFF_P0 = OFF_A0 + (size_t)K * K;          // 16384  P0 = A0^{-1}
  const size_t OFF_AG = OFF_P0 + (size_t)K * K;          // 16384  GJ scratch
  const size_t OFF_P  = OFF_AG + (size_t)K * K;          // B*K*K  per-batch P
  const size_t OFF_U  = OFF_P + (size_t)B * K * K;       // B*K*K  per-batch U
  const size_t OFF_KL = OFF_U + (size_t)B * K * K;       // 2*E*B  kl slots

  gram_kernel<<<64, 32, 0, stream>>>(M0, ws + OFF_A0);
  invert_kernel<<<1, 256, 0, stream>>>(ws + OFF_A0, ws + OFF_AG, ws + OFF_P0);
  scan_kernel<<<B, 512, 0, stream>>>(x, M0, pvar, plw, zn, out,
                                     ws + OFF_P, ws + OFF_U,
                                     ws + OFF_P0, ws + OFF_KL);
  kl_reduce_kernel<<<1, 256, 0, stream>>>(ws + OFF_KL, out + (size_t)B * E * C);
}